// LatentSDE_5093831213857
// MI455X (gfx1250) — compile-verified
//
#include <hip/hip_runtime.h>
#include <math.h>

// Problem constants (from reference): N=45, B=32768, HID=64
#define NT       45
#define BB       32768
#define ROWS     (NT * BB)      // 1474560 flattened (n,b) rows
#define CHUNKS   (ROWS / 16)    // 92160 row-chunks of 16
#define SUBS     4              // lanes cooperating per batch element in scan
#define K1_BLOCKS 512           // 512*256 = 131072 = 4 lanes per batch elem
#define K2_BLOCKS 1440          // grid-stride: 1440*8 waves -> 8 chunks/wave
#define WAVES2    8             // 256 threads = 8 wave32

#define LOG2E 1.44269504088896340736f
#define LN2   0.69314718055994530942f

typedef __attribute__((ext_vector_type(2))) float v2f;
typedef __attribute__((ext_vector_type(8))) float v8f;

// ---------------------------------------------------------------------------
// Kernel 1: encoder + posterior + 44-step SDE scan.
// 4 lanes cooperate on one batch element: each handles j = 4*jj+sub of the
// 64-wide hidden layers; partials combined with 2 shfl_xor butterflies.
// The z-state update is replicated across the 4 lanes (identical fp ops ->
// identical values -> deterministic). KL/path counted once (sub==0).
// ---------------------------------------------------------------------------
__global__ __launch_bounds__(256) void sde_scan_kernel(
    const float* __restrict__ xs,   const float* __restrict__ ts,
    const float* __restrict__ eps0, const float* __restrict__ bm_eps,
    const float* __restrict__ ew1,  const float* __restrict__ eb1,
    const float* __restrict__ ew2,  const float* __restrict__ eb2,
    const float* __restrict__ qw,   const float* __restrict__ qb,
    const float* __restrict__ fw1,  const float* __restrict__ fb1,
    const float* __restrict__ fw2,  const float* __restrict__ fb2,
    const float* __restrict__ pzm,  const float* __restrict__ pzls,
    const float* __restrict__ kap,  const float* __restrict__ the,
    const float* __restrict__ sig,
    float* __restrict__ zs, float* __restrict__ kl_part,
    float* __restrict__ path_part)
{
    // LDS staging of all small weights (585 floats)
    __shared__ float sh[592];
    const int FW1 = 0, FB1 = 192, FW2 = 256, EW1 = 320, EB1 = 384, EW2 = 448,
              EB2 = 576, QW = 578, QB = 582, FB2 = 584;
    const int tid = threadIdx.x;
    for (int i = tid; i < 192; i += 256) sh[FW1 + i] = fw1[i];
    for (int i = tid; i < 64; i += 256) {
        sh[FB1 + i] = fb1[i];  sh[FW2 + i] = fw2[i];
        sh[EW1 + i] = ew1[i];  sh[EB1 + i] = eb1[i];
    }
    for (int i = tid; i < 128; i += 256) sh[EW2 + i] = ew2[i];
    if (tid < 2) { sh[EB2 + tid] = eb2[tid]; sh[QB + tid] = qb[tid]; }
    if (tid < 4) sh[QW + tid] = qw[tid];
    if (tid == 0) sh[FB2] = fb2[0];
    __syncthreads();

    const float kappa = kap[0], theta = the[0], sigma = sig[0];
    const float pmean = pzm[0], plogstd = pzls[0];

    const int t   = blockIdx.x * 256 + tid;
    const int b   = t >> 2;          // batch element
    const int sub = t & (SUBS - 1);  // lane role within the group of 4

    float z = 0.0f, kl = 0.0f, path = 0.0f, tprev = ts[0];

    for (int n = 0; n < NT; n++) {
        const float x = xs[n * BB + b];
        // encoder MLP 1 -> 64 -> 2, j split across the 4 sub-lanes
        float c0 = 0.0f, c1 = 0.0f;
        #pragma unroll 4
        for (int jj = 0; jj < 16; jj++) {
            const int j = jj * 4 + sub;
            const float h = fmaxf(fmaf(x, sh[EW1 + j], sh[EB1 + j]), 0.0f);
            c0 = fmaf(h, sh[EW2 + 2 * j], c0);
            c1 = fmaf(h, sh[EW2 + 2 * j + 1], c1);
        }
        c0 += __shfl_xor(c0, 1, 32); c0 += __shfl_xor(c0, 2, 32);
        c1 += __shfl_xor(c1, 1, 32); c1 += __shfl_xor(c1, 2, 32);
        c0 += sh[EB2 + 0];
        c1 += sh[EB2 + 1];

        if (n == 0) {
            const float qm  = c0 * sh[QW + 0] + c1 * sh[QW + 2] + sh[QB + 0];
            const float qls = c0 * sh[QW + 1] + c1 * sh[QW + 3] + sh[QB + 1];
            z = qm + expf(qls) * eps0[b];
            const float sq2 = expf(2.0f * qls);
            const float sp2 = expf(2.0f * plogstd);
            const float dm  = qm - pmean;
            kl = plogstd - qls + (sq2 + dm * dm) / (2.0f * sp2) - 0.5f;
            if (sub == 0) zs[b] = z;
        } else {
            const float tn = ts[n];
            const float dt = tn - tprev; tprev = tn;
            const float dw = bm_eps[(n - 1) * BB + b] * sqrtf(dt);
            // f MLP: concat([z, c0, c1]) -> 64 -> 1, split across sub-lanes
            float fp = 0.0f;
            #pragma unroll 4
            for (int jj = 0; jj < 16; jj++) {
                const int j = jj * 4 + sub;
                float h = z * sh[FW1 + j] + c0 * sh[FW1 + 64 + j] +
                          c1 * sh[FW1 + 128 + j] + sh[FB1 + j];
                h = fmaxf(h, 0.0f);
                fp = fmaf(h, sh[FW2 + j], fp);
            }
            fp += __shfl_xor(fp, 1, 32);
            fp += __shfl_xor(fp, 2, 32);
            const float f = fp + sh[FB2];
            const float hd = kappa * (theta - z);
            const float u  = (f - hd) / sigma;
            path += 0.5f * u * u * dt;
            z = z + f * dt + sigma * dw;
            if (sub == 0) zs[n * BB + b] = z;
        }
    }

    // count KL/path once per batch element, then deterministic tree reduce
    if (sub != 0) { kl = 0.0f; path = 0.0f; }
    __shared__ float r1[256], r2[256];
    r1[tid] = kl; r2[tid] = path;
    __syncthreads();
    for (int s = 128; s > 0; s >>= 1) {
        if (tid < s) { r1[tid] += r1[tid + s]; r2[tid] += r2[tid + s]; }
        __syncthreads();
    }
    if (tid == 0) { kl_part[blockIdx.x] = r1[0]; path_part[blockIdx.x] = r2[0]; }
}

// ---------------------------------------------------------------------------
// Kernel 2: projection decoder with fp32 WMMA.
// Per 16-row chunk: h1 = softplus(z*pw1+pb1) built directly in the ISA's
// 16x4 fp32 A-fragment layout, 16 chained V_WMMA_F32_16X16X4_F32 per column
// tile accumulate the 64x64 pw2 GEMM in exact fp32, then relu + pw3 dot.
// pw2 staged in LDS pair-interleaved so each B fragment is one ds_load_b64.
// Softplus uses the native v_exp_f32/v_log_f32 transcendental path.
// ---------------------------------------------------------------------------
__global__ __launch_bounds__(256) void proj_wmma_kernel(
    const float* __restrict__ xs,  const float* __restrict__ zs,
    const float* __restrict__ pw1, const float* __restrict__ pb1,
    const float* __restrict__ pw2, const float* __restrict__ pb2,
    const float* __restrict__ pw3, const float* __restrict__ pb3,
    const float* __restrict__ noise_std,
    float* __restrict__ xs_out,    // d_out + 2, flattened (N,B)
    float* __restrict__ sse_part)
{
    // pw2 pair-interleaved: pairIdx m holds (pw2[2m][col], pw2[2m+1][col])
    __shared__ float w2l[4096];
    __shared__ float redbuf[WAVES2][256];
    const int tid = threadIdx.x;
    for (int i = tid; i < 2048; i += 256) {
        const int m = i >> 6, col = i & 63;
        w2l[2 * i]     = pw2[(2 * m) * 64 + col];
        w2l[2 * i + 1] = pw2[(2 * m + 1) * 64 + col];
    }
    __syncthreads();

    const int lane = tid & 31;
    const int wave = tid >> 5;
    const int hf   = lane >> 4;    // wave half: 0 -> K=4k,4k+1 ; 1 -> K=4k+2,4k+3
    const int l16  = lane & 15;
    const int koff = hf * 2;

    // hoisted per-lane constants (fixed across chunks)
    float w1k[32], b1k[32];
    #pragma unroll
    for (int i = 0; i < 32; i++) {
        const int k = 4 * (i >> 1) + koff + (i & 1);
        w1k[i] = pw1[k]; b1k[i] = pb1[k];
    }
    float bias2[4], w3c[4];
    #pragma unroll
    for (int ct = 0; ct < 4; ct++) {
        bias2[ct] = pb2[16 * ct + l16];
        w3c[ct]   = pw3[16 * ct + l16];
    }
    const float b3 = pb3[0];
    const float inv_std = 1.0f / noise_std[0];

    float sse = 0.0f;
    for (int chunk = blockIdx.x * WAVES2 + wave; chunk < CHUNKS;
         chunk += gridDim.x * WAVES2) {
        const int row0 = chunk * 16;
        const float zv = zs[row0 + l16];   // this lane's row (M = l16)

        // A fragments: softplus(z*pw1+pb1) in WMMA 16x4 fp32 layout.
        // softplus(x) = max(x,0) + ln2 * log2(1 + exp2(-|x| * log2e))
        float af[32];
        #pragma unroll
        for (int i = 0; i < 32; i++) {
            const float p = fmaf(zv, w1k[i], b1k[i]);
            const float e = __builtin_amdgcn_exp2f(-fabsf(p) * LOG2E);
            af[i] = fmaxf(p, 0.0f) + LN2 * __builtin_amdgcn_logf(1.0f + e);
        }

        v8f acc[4];
        #pragma unroll
        for (int ct = 0; ct < 4; ct++)
            #pragma unroll
            for (int v = 0; v < 8; v++) acc[ct][v] = bias2[ct];

        #pragma unroll
        for (int kk = 0; kk < 16; kk++) {
            v2f a; a[0] = af[2 * kk]; a[1] = af[2 * kk + 1];
            const int m = 2 * kk + hf;  // pair index for this half's K rows
            #pragma unroll
            for (int ct = 0; ct < 4; ct++) {
                const v2f bfr =
                    *reinterpret_cast<const v2f*>(&w2l[(m * 64 + 16 * ct + l16) * 2]);
                acc[ct] = __builtin_amdgcn_wmma_f32_16x16x4_f32(
                    false, a, false, bfr, (short)0, acc[ct], false, false);
            }
        }

        // h2 = relu(acc); per-lane partial dot with pw3 columns
        #pragma unroll
        for (int v = 0; v < 8; v++) {
            float s = 0.0f;
            #pragma unroll
            for (int ct = 0; ct < 4; ct++)
                s = fmaf(fmaxf(acc[ct][v], 0.0f), w3c[ct], s);
            const int r = v + hf * 8;          // D layout: M = v + 8*half
            redbuf[wave][r * 16 + l16] = s;
        }
        __builtin_amdgcn_wave_barrier();       // wave-private LDS, DS is in-order
        if (lane < 16) {
            float rs = 0.0f;
            #pragma unroll
            for (int s = 0; s < 16; s++) rs += redbuf[wave][lane * 16 + s];
            const float xo = rs + b3;
            xs_out[row0 + lane] = xo;
            const float e = (xs[row0 + lane] - xo) * inv_std;
            sse += -0.5f * e * e;
        }
        __builtin_amdgcn_wave_barrier();
    }

    // deterministic block tree reduction of SSE
    __shared__ float rb[256];
    rb[tid] = sse;
    __syncthreads();
    for (int s = 128; s > 0; s >>= 1) {
        if (tid < s) rb[tid] += rb[tid + s];
        __syncthreads();
    }
    if (tid == 0) sse_part[blockIdx.x] = rb[0];
}

// ---------------------------------------------------------------------------
// Kernel 3: serial finalize of the two scalar outputs (deterministic order).
// ---------------------------------------------------------------------------
__global__ void finalize_kernel(const float* __restrict__ kl_part,
                                const float* __restrict__ path_part,
                                const float* __restrict__ sse_part,
                                const float* __restrict__ noise_std,
                                float* __restrict__ out)
{
    float kl = 0.0f, path = 0.0f, sse = 0.0f;
    for (int i = 0; i < K1_BLOCKS; i++) { kl += kl_part[i]; path += path_part[i]; }
    for (int i = 0; i < K2_BLOCKS; i++) sse += sse_part[i];
    const float std = noise_std[0];
    const float log2pi = 1.8378770664093453f;  // log(2*pi)
    out[0] = sse / (float)BB + (float)NT * (-logf(std) - 0.5f * log2pi);
    out[1] = (kl + path) / (float)BB;
}

// ---------------------------------------------------------------------------
extern "C" void kernel_launch(void* const* d_in, const int* in_sizes, int n_in,
                              void* d_out, int out_size, void* d_ws, size_t ws_size,
                              hipStream_t stream)
{
    const float* xs        = (const float*)d_in[0];
    const float* ts        = (const float*)d_in[1];
    const float* noise_std = (const float*)d_in[2];
    const float* eps0      = (const float*)d_in[3];
    const float* bm_eps    = (const float*)d_in[4];
    const float* enc_w1    = (const float*)d_in[5];
    const float* enc_b1    = (const float*)d_in[6];
    const float* enc_w2    = (const float*)d_in[7];
    const float* enc_b2    = (const float*)d_in[8];
    const float* qz0_w     = (const float*)d_in[9];
    const float* qz0_b     = (const float*)d_in[10];
    const float* f_w1      = (const float*)d_in[11];
    const float* f_b1      = (const float*)d_in[12];
    const float* f_w2      = (const float*)d_in[13];
    const float* f_b2      = (const float*)d_in[14];
    const float* proj_w1   = (const float*)d_in[15];
    const float* proj_b1   = (const float*)d_in[16];
    const float* proj_w2   = (const float*)d_in[17];
    const float* proj_b2   = (const float*)d_in[18];
    const float* proj_w3   = (const float*)d_in[19];
    const float* proj_b3   = (const float*)d_in[20];
    const float* pz0_mean  = (const float*)d_in[21];
    const float* pz0_lstd  = (const float*)d_in[22];
    const float* kappa     = (const float*)d_in[23];
    const float* theta     = (const float*)d_in[24];
    const float* sigma     = (const float*)d_in[25];

    float* out = (float*)d_out;             // [log_pxs, logqp, _xs(N*B)]
    float* ws  = (float*)d_ws;
    float* zs        = ws;                  // N*B
    float* kl_part   = ws + ROWS;           // K1_BLOCKS
    float* path_part = kl_part + K1_BLOCKS; // K1_BLOCKS
    float* sse_part  = path_part + K1_BLOCKS; // K2_BLOCKS

    sde_scan_kernel<<<K1_BLOCKS, 256, 0, stream>>>(
        xs, ts, eps0, bm_eps, enc_w1, enc_b1, enc_w2, enc_b2,
        qz0_w, qz0_b, f_w1, f_b1, f_w2, f_b2,
        pz0_mean, pz0_lstd, kappa, theta, sigma,
        zs, kl_part, path_part);

    proj_wmma_kernel<<<K2_BLOCKS, 256, 0, stream>>>(
        xs, zs, proj_w1, proj_b1, proj_w2, proj_b2, proj_w3, proj_b3,
        noise_std, out + 2, sse_part);

    finalize_kernel<<<1, 1, 0, stream>>>(kl_part, path_part, sse_part,
                                         noise_std, out);
}